// CombinedLoss_19980187861539
// MI455X (gfx1250) — compile-verified
//
#include <hip/hip_runtime.h>
#include <hip/hip_bf16.h>

// Sinkhorn EMD for B=16, N=2048, D=3, eps=0.01, 50 iterations.
// K stored as bf16 (128 MB) -> resident in MI455X's 192 MB L2; the 100
// matvec passes (12.8 GB) then run at L2 bandwidth, not HBM (23.3 TB/s).
// bf16 keeps fp32's exponent range (K spans e^0..e^-87; fp16 would flush).
// WMMA f32 16x16x4 computes the x*y^T Gram tiles in the build kernel.
// Column passes use b128 loads (8 cols/thread) so the L2-resident streaming
// is issue-rate efficient; vector staging uses async global->LDS when the
// toolchain exposes the gfx1250 builtins.

#define BATCH 16
#define NPTS  2048
#define SINK_ITERS 50
#define NCHUNK 16
#define CHROWS (NPTS / NCHUNK)          // 128 rows per column-pass chunk
#define RCPN (1.0f / 2048.0f)
// exp(-C/eps) = exp2(-C * log2(e)/eps), eps = 0.01
#define NEG_LOG2E_OVER_EPS (-144.2695040889f)

typedef float v2f __attribute__((ext_vector_type(2)));
typedef float v8f __attribute__((ext_vector_type(8)));

typedef __attribute__((address_space(1))) int* gptr_i32;
typedef __attribute__((address_space(3))) int* lptr_i32;

static __device__ __forceinline__ float bf2f(unsigned short h) {
    union { unsigned u; float f; } c; c.u = ((unsigned)h) << 16; return c.f;
}
static __device__ __forceinline__ float bfLo(unsigned w) {
    union { unsigned u; float f; } c; c.u = w << 16; return c.f;
}
static __device__ __forceinline__ float bfHi(unsigned w) {
    union { unsigned u; float f; } c; c.u = w & 0xFFFF0000u; return c.f;
}
static __device__ __forceinline__ unsigned short f2bf(float f) {
    union { float f; unsigned u; } c; c.f = f;
    unsigned r = c.u + 0x7FFFu + ((c.u >> 16) & 1u);   // round to nearest even
    return (unsigned short)(r >> 16);
}

// ---- async global->LDS staging (gfx1250 GLOBAL_LOAD_ASYNC_TO_LDS_B32) ----
#if defined(__gfx1250__) && \
    __has_builtin(__builtin_amdgcn_global_load_async_to_lds_b32) && \
    __has_builtin(__builtin_amdgcn_s_wait_asynccnt)
#define USE_ASYNC_LDS 1
#endif

static __device__ __forceinline__ void stage_b32(const float* g, float* l) {
#ifdef USE_ASYNC_LDS
    __builtin_amdgcn_global_load_async_to_lds_b32(
        (gptr_i32)(uintptr_t)g,
        (lptr_i32)(unsigned)(uintptr_t)l,
        0, 0);
#else
    *l = *g;
#endif
}
static __device__ __forceinline__ void stage_wait_and_sync() {
#ifdef USE_ASYNC_LDS
    __builtin_amdgcn_s_wait_asynccnt(0);
#endif
    __syncthreads();
}

// ---------------------------------------------------------------------------
// Build K[b,i,j] = exp(-||x_i - y_j|| / eps) as bf16.
// Each wave computes one 16x16 tile; the Gram term x.y^T comes from a single
// v_wmma_f32_16x16x4_f32 (D=3 padded to K=4 with zeros).
// ---------------------------------------------------------------------------
__global__ void __launch_bounds__(256)
build_K_kernel(const float* __restrict__ x, const float* __restrict__ y,
               unsigned short* __restrict__ K)
{
    __shared__ float sx2[8][16];
    __shared__ float sy2[8][16];
    const int b    = blockIdx.z;
    const int m0   = blockIdx.y * 16;
    const int wave = threadIdx.x >> 5;
    const int lane = threadIdx.x & 31;
    const int half = lane >> 4;
    const int l16  = lane & 15;
    const int n0   = blockIdx.x * 128 + wave * 16;

    const float* xr = x + ((size_t)b * NPTS + (m0 + l16)) * 3;
    const float* yr = y + ((size_t)b * NPTS + (n0 + l16)) * 3;
    const float x0 = xr[0], x1 = xr[1], x2 = xr[2];
    const float y0 = yr[0], y1 = yr[1], y2 = yr[2];

    if (half == 0) {
        sx2[wave][l16] = x0 * x0 + x1 * x1 + x2 * x2;
        sy2[wave][l16] = y0 * y0 + y1 * y1 + y2 * y2;
    }

    v2f a, bb;
    a.x  = half ? x2 : x0;  a.y  = half ? 0.0f : x1;
    bb.x = half ? y2 : y0;  bb.y = half ? 0.0f : y1;

    v8f acc = {0.f, 0.f, 0.f, 0.f, 0.f, 0.f, 0.f, 0.f};
    acc = __builtin_amdgcn_wmma_f32_16x16x4_f32(
        /*neg_a=*/false, a, /*neg_b=*/false, bb,
        /*c_mod=*/(short)0, acc, /*reuse_a=*/false, /*reuse_b=*/false);

    const float y2n = sy2[wave][l16];
    unsigned short* Kout =
        K + ((size_t)b * NPTS + (size_t)(m0 + half * 8)) * NPTS + (n0 + l16);
#pragma unroll
    for (int r = 0; r < 8; ++r) {
        float xy  = acc[r];
        float x2m = sx2[wave][half * 8 + r];
        float d2  = fmaxf(x2m + y2n - 2.0f * xy, 1e-12f);
        float Cc  = __builtin_sqrtf(d2);
        float Kv  = __builtin_exp2f(Cc * NEG_LOG2E_OVER_EPS);
        Kout[(size_t)r * NPTS] = f2bf(Kv);
    }
}

// ---------------------------------------------------------------------------
// Column partial sums of K over CHROWS-row chunks (deterministic, no
// atomics). grid = (NCHUNK, B). A 256-thread block covers all 2048 columns:
// each thread owns 8 consecutive columns via one b128 load per row.
// ---------------------------------------------------------------------------
__global__ void __launch_bounds__(256)
colsum_part_kernel(const unsigned short* __restrict__ K, float* __restrict__ part)
{
    const int b  = blockIdx.y;
    const int ch = blockIdx.x;
    const int j  = threadIdx.x * 8;
    const unsigned short* Kp =
        K + (size_t)b * NPTS * NPTS + (size_t)(ch * CHROWS) * NPTS + j;
    float a0 = 0.f, a1 = 0.f, a2 = 0.f, a3 = 0.f;
    float a4 = 0.f, a5 = 0.f, a6 = 0.f, a7 = 0.f;
    for (int i0 = 0; i0 < CHROWS; i0 += 8) {
        __builtin_prefetch((const void*)(Kp + (size_t)16 * NPTS), 0, 0);
#pragma unroll
        for (int i = 0; i < 8; ++i) {
            uint4 pk = *(const uint4*)Kp;
            a0 += bfLo(pk.x); a1 += bfHi(pk.x);
            a2 += bfLo(pk.y); a3 += bfHi(pk.y);
            a4 += bfLo(pk.z); a5 += bfHi(pk.z);
            a6 += bfLo(pk.w); a7 += bfHi(pk.w);
            Kp += NPTS;
        }
    }
    float* po = part + ((size_t)(b * NCHUNK + ch)) * NPTS + j;
    *(float4*)(po + 0) = make_float4(a0, a1, a2, a3);
    *(float4*)(po + 4) = make_float4(a4, a5, a6, a7);
}

// ---------------------------------------------------------------------------
// Combine NCHUNK chunk partials per column (fixed order -> deterministic).
// mode 0: s[j] = colsum, vp[j] = 1/s   (initial v=1 folded: vp = v/s)
// mode 1: vp[j] = (1/N) / (K^T u)_j    (v-update, s folded out of K_tilde)
// ---------------------------------------------------------------------------
__global__ void __launch_bounds__(256)
combine_kernel(const float* __restrict__ part, float* __restrict__ s,
               float* __restrict__ vp, int mode)
{
    const int b = blockIdx.y;
    const int j = blockIdx.x * 256 + threadIdx.x;
    float sum = 0.f;
#pragma unroll
    for (int ch = 0; ch < NCHUNK; ++ch)
        sum += part[((size_t)(b * NCHUNK + ch)) * NPTS + j];
    if (mode == 0) {
        s[(size_t)b * NPTS + j]  = sum;
        vp[(size_t)b * NPTS + j] = 1.0f / sum;
    } else {
        vp[(size_t)b * NPTS + j] = RCPN / sum;
    }
}

// ---------------------------------------------------------------------------
// u_i = (1/N) / sum_j K_ij * vp_j. One wave per row, vp staged in LDS
// (async global->LDS when available), K rows read as b128.
// ---------------------------------------------------------------------------
__global__ void __launch_bounds__(256)
u_kernel(const unsigned short* __restrict__ K, const float* __restrict__ vp,
         float* __restrict__ u)
{
    __shared__ float svp[NPTS];
    const int b = blockIdx.y;
    for (int t = threadIdx.x; t < NPTS; t += 256)
        stage_b32(vp + (size_t)b * NPTS + t, &svp[t]);
    stage_wait_and_sync();

    const int wave = threadIdx.x >> 5;
    const int lane = threadIdx.x & 31;
    const int row  = blockIdx.x * 8 + wave;
    const unsigned short* Kr = K + ((size_t)b * NPTS + row) * NPTS;

    float acc = 0.f;
    for (int j0 = lane * 8; j0 < NPTS; j0 += 256) {
        uint4 pk = *(const uint4*)(Kr + j0);
        acc += bfLo(pk.x) * svp[j0 + 0];
        acc += bfHi(pk.x) * svp[j0 + 1];
        acc += bfLo(pk.y) * svp[j0 + 2];
        acc += bfHi(pk.y) * svp[j0 + 3];
        acc += bfLo(pk.z) * svp[j0 + 4];
        acc += bfHi(pk.z) * svp[j0 + 5];
        acc += bfLo(pk.w) * svp[j0 + 6];
        acc += bfHi(pk.w) * svp[j0 + 7];
    }
#pragma unroll
    for (int off = 16; off > 0; off >>= 1)
        acc += __shfl_down(acc, off, 32);
    if (lane == 0)
        u[(size_t)b * NPTS + row] = RCPN / acc;
}

// ---------------------------------------------------------------------------
// Column dot (K^T u) partials over CHROWS-row chunks; u chunk staged in LDS
// (async). Same full-row-width b128 layout as colsum_part_kernel.
// ---------------------------------------------------------------------------
__global__ void __launch_bounds__(256)
vdot_part_kernel(const unsigned short* __restrict__ K, const float* __restrict__ u,
                 float* __restrict__ part)
{
    __shared__ float su[CHROWS];
    const int b  = blockIdx.y;
    const int ch = blockIdx.x;
    if (threadIdx.x < CHROWS)
        stage_b32(u + (size_t)b * NPTS + ch * CHROWS + threadIdx.x,
                  &su[threadIdx.x]);
    stage_wait_and_sync();

    const int j = threadIdx.x * 8;
    const unsigned short* Kp =
        K + (size_t)b * NPTS * NPTS + (size_t)(ch * CHROWS) * NPTS + j;
    float a0 = 0.f, a1 = 0.f, a2 = 0.f, a3 = 0.f;
    float a4 = 0.f, a5 = 0.f, a6 = 0.f, a7 = 0.f;
    for (int i0 = 0; i0 < CHROWS; i0 += 8) {
        __builtin_prefetch((const void*)(Kp + (size_t)16 * NPTS), 0, 0);
#pragma unroll
        for (int i = 0; i < 8; ++i) {
            uint4 pk = *(const uint4*)Kp;
            float uu = su[i0 + i];
            a0 += bfLo(pk.x) * uu; a1 += bfHi(pk.x) * uu;
            a2 += bfLo(pk.y) * uu; a3 += bfHi(pk.y) * uu;
            a4 += bfLo(pk.z) * uu; a5 += bfHi(pk.z) * uu;
            a6 += bfLo(pk.w) * uu; a7 += bfHi(pk.w) * uu;
            Kp += NPTS;
        }
    }
    float* po = part + ((size_t)(b * NCHUNK + ch)) * NPTS + j;
    *(float4*)(po + 0) = make_float4(a0, a1, a2, a3);
    *(float4*)(po + 4) = make_float4(a4, a5, a6, a7);
}

// ---------------------------------------------------------------------------
// loss partials: sum_ij u_i * K_ij * v_j * C_ij, with v_j = vp_j * s_j and
// C_ij recomputed from coordinates (y tile + v staged in LDS: 32 KB).
// One wave per row; one partial per block (no atomics).
// ---------------------------------------------------------------------------
__global__ void __launch_bounds__(256)
loss_part_kernel(const float* __restrict__ x, const float* __restrict__ y,
                 const unsigned short* __restrict__ K,
                 const float* __restrict__ u, const float* __restrict__ vp,
                 const float* __restrict__ s, float* __restrict__ part)
{
    __shared__ float sv[NPTS];
    __shared__ float sy[NPTS * 3];
    __shared__ float wred[8];
    const int b = blockIdx.y;
    for (int t = threadIdx.x; t < NPTS; t += 256)
        sv[t] = vp[(size_t)b * NPTS + t] * s[(size_t)b * NPTS + t];
    for (int t = threadIdx.x; t < NPTS * 3; t += 256)
        stage_b32(y + (size_t)b * NPTS * 3 + t, &sy[t]);
    stage_wait_and_sync();

    const int wave = threadIdx.x >> 5;
    const int lane = threadIdx.x & 31;
    const int row  = blockIdx.x * 8 + wave;
    const float* xr = x + ((size_t)b * NPTS + row) * 3;
    const float x0 = xr[0], x1 = xr[1], x2 = xr[2];
    const float ui = u[(size_t)b * NPTS + row];
    const unsigned short* Kr = K + ((size_t)b * NPTS + row) * NPTS;

    float acc = 0.f;
    for (int j = lane; j < NPTS; j += 32) {
        float Kv = bf2f(Kr[j]);
        float dx = x0 - sy[3 * j + 0];
        float dy = x1 - sy[3 * j + 1];
        float dz = x2 - sy[3 * j + 2];
        float d2 = fmaxf(dx * dx + dy * dy + dz * dz, 1e-12f);
        acc += Kv * sv[j] * __builtin_sqrtf(d2);
    }
    acc *= ui;
#pragma unroll
    for (int off = 16; off > 0; off >>= 1)
        acc += __shfl_down(acc, off, 32);
    if (lane == 0) wred[wave] = acc;
    __syncthreads();
    if (threadIdx.x == 0) {
        float t = 0.f;
#pragma unroll
        for (int w = 0; w < 8; ++w) t += wred[w];
        part[(size_t)b * gridDim.x + blockIdx.x] = t;
    }
}

// Deterministic final reduction of 4096 block partials; writes mean over B.
__global__ void __launch_bounds__(256)
final_reduce_kernel(const float* __restrict__ part, float* __restrict__ out)
{
    __shared__ float red[256];
    float acc = 0.f;
    for (int i = threadIdx.x; i < 256 * BATCH; i += 256) acc += part[i];
    red[threadIdx.x] = acc;
    __syncthreads();
    for (int st = 128; st > 0; st >>= 1) {
        if ((int)threadIdx.x < st) red[threadIdx.x] += red[threadIdx.x + st];
        __syncthreads();
    }
    if (threadIdx.x == 0) out[0] = red[0] * (1.0f / BATCH);
}

extern "C" void kernel_launch(void* const* d_in, const int* in_sizes, int n_in,
                              void* d_out, int out_size, void* d_ws, size_t ws_size,
                              hipStream_t stream)
{
    (void)in_sizes; (void)n_in; (void)out_size; (void)ws_size;
    const float* pc1 = (const float*)d_in[0];   // predicted cloud (x)
    const float* pc2 = (const float*)d_in[1];   // ground truth  (y)
    // pc3 (d_in[2]) is unused by the loss.

    char* ws = (char*)d_ws;
    const size_t szK = (size_t)BATCH * NPTS * NPTS * sizeof(unsigned short); // 128 MB
    unsigned short* K = (unsigned short*)ws;
    float* s    = (float*)(ws + szK);
    float* u    = s  + (size_t)BATCH * NPTS;
    float* vp   = u  + (size_t)BATCH * NPTS;
    float* part = vp + (size_t)BATCH * NPTS;   // NCHUNK*B*N floats = 2 MB

    const dim3 blk(256);

    build_K_kernel<<<dim3(NPTS / 128, NPTS / 16, BATCH), blk, 0, stream>>>(pc1, pc2, K);

    colsum_part_kernel<<<dim3(NCHUNK, BATCH), blk, 0, stream>>>(K, part);
    combine_kernel<<<dim3(NPTS / 256, BATCH), blk, 0, stream>>>(part, s, vp, 0);

    for (int it = 0; it < SINK_ITERS; ++it) {
        u_kernel<<<dim3(NPTS / 8, BATCH), blk, 0, stream>>>(K, vp, u);
        vdot_part_kernel<<<dim3(NCHUNK, BATCH), blk, 0, stream>>>(K, u, part);
        combine_kernel<<<dim3(NPTS / 256, BATCH), blk, 0, stream>>>(part, s, vp, 1);
    }

    loss_part_kernel<<<dim3(NPTS / 8, BATCH), blk, 0, stream>>>(pc1, pc2, K, u, vp, s, part);
    final_reduce_kernel<<<1, blk, 0, stream>>>(part, (float*)d_out);
}